// GraphConvNet_55052890800550
// MI455X (gfx1250) — compile-verified
//
#include <hip/hip_runtime.h>
#include <math.h>

typedef float v2f __attribute__((ext_vector_type(2)));
typedef float v8f __attribute__((ext_vector_type(8)));

// ---------------------------------------------------------------- zero fill
__global__ __launch_bounds__(256) void zero_k(float* __restrict__ p, long n) {
    long i = (long)blockIdx.x * blockDim.x + threadIdx.x;
    long stride = (long)gridDim.x * blockDim.x;
    for (; i < n; i += stride) p[i] = 0.0f;
}

// -------------------------------------------- weight repack: row-pair interleave
// wp[k2][col][j] = w[2*k2 + j][col]   (j in {0,1})
// Makes a WMMA B-fragment (K = vgpr + 2*lh) a single aligned b64 load.
__global__ __launch_bounds__(256)
void repack_w_k(const float* __restrict__ w, float* __restrict__ wp,
                int d_in, int d_out) {
    int t = blockIdx.x * blockDim.x + threadIdx.x;
    int total = (d_in / 2) * d_out;
    if (t >= total) return;
    int k2  = t / d_out;
    int col = t % d_out;
    wp[(size_t)k2 * d_out * 2 + col * 2 + 0] = w[(size_t)(2 * k2 + 0) * d_out + col];
    wp[(size_t)k2 * d_out * 2 + col * 2 + 1] = w[(size_t)(2 * k2 + 1) * d_out + col];
}

// ------------------------------------------------- edge scatter-add (agg)
// One wave32 per edge: 32 lanes x float4 = 128 floats per pass.
template <int D>
__global__ __launch_bounds__(256)
void scatter_add_k(const float* __restrict__ xin, const int* __restrict__ src,
                   const int* __restrict__ dst, float* __restrict__ agg, int n_edges) {
    int e    = blockIdx.x * 8 + (threadIdx.x >> 5);
    int lane = threadIdx.x & 31;
    if (e >= n_edges) return;
    int s = src[e];
    int d = dst[e];
    const float* xs = xin + (size_t)s * D;
    float*       ad = agg + (size_t)d * D;
#pragma unroll
    for (int it = 0; it < D / 128; ++it) {
        int o = it * 128 + lane * 4;
        float4 v = *(const float4*)(xs + o);
        __hip_atomic_fetch_add(ad + o + 0, v.x, __ATOMIC_RELAXED, __HIP_MEMORY_SCOPE_AGENT);
        __hip_atomic_fetch_add(ad + o + 1, v.y, __ATOMIC_RELAXED, __HIP_MEMORY_SCOPE_AGENT);
        __hip_atomic_fetch_add(ad + o + 2, v.z, __ATOMIC_RELAXED, __HIP_MEMORY_SCOPE_AGENT);
        __hip_atomic_fetch_add(ad + o + 3, v.w, __ATOMIC_RELAXED, __HIP_MEMORY_SCOPE_AGENT);
    }
}

// --------------------------------------------- fused dual-GEMM + bias + ELU
// out[ROWS x D_OUT] = elu(agg@w_rel + x@w_root + b), FP32 WMMA 16x16x4.
// Block = 256 threads = 8 waves; each wave owns ROWS/16 stacked 16x16 tiles
// of one 16-column strip, sharing the same B fragments across row tiles.
template <int D_IN, int D_OUT, int ROWS>
__global__ __launch_bounds__(256)
void gemm_dual_elu_k(const float* __restrict__ agg, const float* __restrict__ xin,
                     const float* __restrict__ wp_rel, const float* __restrict__ wp_root,
                     const float* __restrict__ bias, float* __restrict__ out, int n_rows) {
    constexpr int LDA    = D_IN + 4;   // bank-stride 4 over M lanes -> conflict-free
    constexpr int NTILES = ROWS / 16;
    __shared__ float sAgg[ROWS * LDA];
    __shared__ float sX[ROWS * LDA];

    const int row0 = blockIdx.x * ROWS;
    const int tid  = threadIdx.x;

    // Cooperative coalesced stage of both A slabs (float4).
    constexpr int NF4 = ROWS * D_IN / 4;
    for (int idx = tid; idx < NF4; idx += 256) {
        int r  = idx / (D_IN / 4);
        int c4 = (idx % (D_IN / 4)) * 4;
        int gr = row0 + r;
        float4 va = make_float4(0.f, 0.f, 0.f, 0.f);
        float4 vx = make_float4(0.f, 0.f, 0.f, 0.f);
        if (gr < n_rows) {
            va = *(const float4*)(agg + (size_t)gr * D_IN + c4);
            vx = *(const float4*)(xin + (size_t)gr * D_IN + c4);
        }
        *(float4*)(&sAgg[r * LDA + c4]) = va;
        *(float4*)(&sX[r * LDA + c4])   = vx;
    }
    __syncthreads();

    const int wave = tid >> 5;
    const int lane = tid & 31;
    const int lm   = lane & 15;   // M (A/C) or N (B/C)
    const int lh   = lane >> 4;   // lane-half selector
    const int col  = blockIdx.y * 128 + wave * 16 + lm;

    const float bv = bias[col];
    v8f c[NTILES];
#pragma unroll
    for (int t = 0; t < NTILES; ++t) c[t] = (v8f){bv, bv, bv, bv, bv, bv, bv, bv};

    for (int k = 0; k < D_IN; k += 4) {
        const int kk = k + 2 * lh;        // K = vgpr_idx + 2*(lane/16)
        const int k2 = (k >> 1) + lh;     // repacked row-pair index

        // Single b64 per matrix per k-step (repacked layout).
        const v2f b1 = *(const v2f*)(&wp_rel[(size_t)k2 * D_OUT * 2 + col * 2]);
        const v2f b2 = *(const v2f*)(&wp_root[(size_t)k2 * D_OUT * 2 + col * 2]);

#pragma unroll
        for (int t = 0; t < NTILES; ++t) {
            const v2f a1 = *(const v2f*)(&sAgg[(t * 16 + lm) * LDA + kk]);
            c[t] = __builtin_amdgcn_wmma_f32_16x16x4_f32(false, a1, false, b1,
                                                         (short)0, c[t], false, false);
            const v2f a2 = *(const v2f*)(&sX[(t * 16 + lm) * LDA + kk]);
            c[t] = __builtin_amdgcn_wmma_f32_16x16x4_f32(false, a2, false, b2,
                                                         (short)0, c[t], false, false);
        }
    }

#pragma unroll
    for (int t = 0; t < NTILES; ++t) {
#pragma unroll
        for (int r = 0; r < 8; ++r) {
            int grow = row0 + t * 16 + r + 8 * lh;  // C/D layout: M = r + 8*(lane/16)
            if (grow < n_rows) {
                float v = c[t][r];
                v = v > 0.0f ? v : (expf(v) - 1.0f);  // ELU
                out[(size_t)grow * D_OUT + col] = v;
            }
        }
    }
}

// ------------------------------------------------------------- graph pooling
__global__ __launch_bounds__(256)
void pool_sum_k(const float* __restrict__ h, const int* __restrict__ batch,
                float* __restrict__ out, int n_nodes) {
    long t     = (long)blockIdx.x * blockDim.x + threadIdx.x;
    long total = (long)n_nodes * 128;
    if (t >= total) return;
    int i = (int)(t >> 7);
    int c = (int)(t & 127);
    int g = batch[i];
    __hip_atomic_fetch_add(&out[g * 128 + c], h[t], __ATOMIC_RELAXED,
                           __HIP_MEMORY_SCOPE_AGENT);
}

__global__ __launch_bounds__(256)
void pool_cnt_k(const int* __restrict__ batch, float* __restrict__ cnt, int n_nodes) {
    int i = blockIdx.x * blockDim.x + threadIdx.x;
    if (i < n_nodes)
        __hip_atomic_fetch_add(&cnt[batch[i]], 1.0f, __ATOMIC_RELAXED,
                               __HIP_MEMORY_SCOPE_AGENT);
}

__global__ __launch_bounds__(256)
void finalize_k(float* __restrict__ out, const float* __restrict__ cnt) {
    int j = blockIdx.x * blockDim.x + threadIdx.x;
    if (j < 64 * 128) out[j] /= fmaxf(cnt[j >> 7], 1.0f);
}

// ------------------------------------------------------------------ driver
extern "C" void kernel_launch(void* const* d_in, const int* in_sizes, int n_in,
                              void* d_out, int out_size, void* d_ws, size_t ws_size,
                              hipStream_t stream) {
    const float* x       = (const float*)d_in[0];
    const int*   ei      = (const int*)d_in[1];
    const int*   batch   = (const int*)d_in[2];
    const float* w_rel1  = (const float*)d_in[3];
    const float* b_rel1  = (const float*)d_in[4];
    const float* w_root1 = (const float*)d_in[5];
    const float* w_rel2  = (const float*)d_in[6];
    const float* b_rel2  = (const float*)d_in[7];
    const float* w_root2 = (const float*)d_in[8];
    const float* w_rel3  = (const float*)d_in[9];
    const float* b_rel3  = (const float*)d_in[10];
    const float* w_root3 = (const float*)d_in[11];

    const int n_nodes = in_sizes[0] / 128;
    const int n_edges = in_sizes[1] / 2;
    const int* src = ei;
    const int* dst = ei + n_edges;

    float* ws  = (float*)d_ws;
    float* agg = ws;                                 // n_nodes*256
    float* h1  = agg + (size_t)n_nodes * 256;        // n_nodes*128
    float* h2  = h1 + (size_t)n_nodes * 128;         // n_nodes*256
    float* h3  = h2 + (size_t)n_nodes * 256;         // n_nodes*128
    float* cnt = h3 + (size_t)n_nodes * 128;         // 64
    float* wp  = cnt + 64;
    float* wp_rel1  = wp;                    // 128*128
    float* wp_root1 = wp_rel1 + 128 * 128;   // 128*128
    float* wp_rel2  = wp_root1 + 128 * 128;  // 128*256
    float* wp_root2 = wp_rel2 + 128 * 256;   // 128*256
    float* wp_rel3  = wp_root2 + 128 * 256;  // 256*128
    float* wp_root3 = wp_rel3 + 256 * 128;   // 256*128
    float* outf = (float*)d_out;

    const int edgeBlocks = (n_edges + 7) / 8;
    const int rt32 = (n_nodes + 31) / 32;
    const int rt16 = (n_nodes + 15) / 16;

    auto zeroN = [&](float* p, long n) {
        int blocks = (int)((n + 255) / 256);
        if (blocks > 4096) blocks = 4096;
        zero_k<<<blocks, 256, 0, stream>>>(p, n);
    };
    auto repack = [&](const float* w, float* wpd, int di, int dο) {
        int total = (di / 2) * dο;
        repack_w_k<<<(total + 255) / 256, 256, 0, stream>>>(w, wpd, di, dο);
    };

    // ---- repack all weights (cheap, deterministic, once per launch)
    repack(w_rel1, wp_rel1, 128, 128);
    repack(w_root1, wp_root1, 128, 128);
    repack(w_rel2, wp_rel2, 128, 256);
    repack(w_root2, wp_root2, 128, 256);
    repack(w_rel3, wp_rel3, 256, 128);
    repack(w_root3, wp_root3, 256, 128);

    // ---- Layer 1: d 128 -> 128
    zeroN(agg, (long)n_nodes * 128);
    scatter_add_k<128><<<edgeBlocks, 256, 0, stream>>>(x, src, dst, agg, n_edges);
    gemm_dual_elu_k<128, 128, 32><<<dim3(rt32, 1), 256, 0, stream>>>(
        agg, x, wp_rel1, wp_root1, b_rel1, h1, n_nodes);

    // ---- Layer 2: d 128 -> 256
    zeroN(agg, (long)n_nodes * 128);
    scatter_add_k<128><<<edgeBlocks, 256, 0, stream>>>(h1, src, dst, agg, n_edges);
    gemm_dual_elu_k<128, 256, 32><<<dim3(rt32, 2), 256, 0, stream>>>(
        agg, h1, wp_rel2, wp_root2, b_rel2, h2, n_nodes);

    // ---- Layer 3: d 256 -> 128
    zeroN(agg, (long)n_nodes * 256);
    scatter_add_k<256><<<edgeBlocks, 256, 0, stream>>>(h2, src, dst, agg, n_edges);
    gemm_dual_elu_k<256, 128, 16><<<dim3(rt16, 1), 256, 0, stream>>>(
        agg, h2, wp_rel3, wp_root3, b_rel3, h3, n_nodes);

    // ---- scatter_mean over graphs
    zeroN(outf, 64 * 128);
    zeroN(cnt, 64);
    long total = (long)n_nodes * 128;
    pool_sum_k<<<(int)((total + 255) / 256), 256, 0, stream>>>(h3, batch, outf, n_nodes);
    pool_cnt_k<<<(n_nodes + 255) / 256, 256, 0, stream>>>(batch, cnt, n_nodes);
    finalize_k<<<(64 * 128 + 255) / 256, 256, 0, stream>>>(outf, cnt);
}